// BiGaBP_unfolding_84731114815618
// MI455X (gfx1250) — compile-verified
//
#include <hip/hip_runtime.h>
#include <math.h>

#define DEV __device__ __forceinline__

typedef float v2f __attribute__((ext_vector_type(2)));
typedef float v8f __attribute__((ext_vector_type(8)));

#if defined(__gfx1250__) && __has_builtin(__builtin_amdgcn_wmma_f32_16x16x4_f32)
#define HAVE_WMMA4 1
#endif
#if defined(__gfx1250__) && __has_builtin(__builtin_amdgcn_permlane16) && \
    __has_builtin(__builtin_amdgcn_permlanex16)
#define HAVE_PERMLANE 1
#endif

namespace {

constexpr int  Bc = 64, Nc = 16, Mc = 16, Kc = 256;
constexpr long SP = (long)Bc * Nc * Mc * Kc;              // 4,194,304 floats per state array
constexpr long OFF_HR = 0 * SP, OFF_HI = 1 * SP, OFF_XR = 2 * SP, OFF_XI = 3 * SP,
               OFF_VX = 4 * SP, OFF_VH = 5 * SP;
constexpr long OFF_PART = 6 * SP;                         // [B][32 chunks][256 (n,m)][3]
constexpr long PART_LEN = (long)Bc * 32 * 256 * 3;        // 1,572,864
constexpr long OFF_KS = OFF_PART + PART_LEN;              // [B][256][3] = 49,152
constexpr long KS_LEN = (long)Bc * 256 * 3;

constexpr long OX  = 32768;                                // out: Hp (B,N,M) cplx -> 32768 floats
constexpr long OVX = 32768 + 524288;                       // then Xp (B,M,K) cplx
constexpr long OVH = 32768 + 524288 + 262144;              // then vXp, then vHp

constexpr float Sq = 0.70710678118654752440f;              // 1/sqrt(2)

DEV float sgnf(float x) { return (x > 0.f) ? 1.f : ((x < 0.f) ? -1.f : 0.f); }

// value from lane ^ 16 (cross-half), VALU permlanex16 when available
DEV float swapx16(float x)
{
#ifdef HAVE_PERMLANE
  int v = __builtin_amdgcn_permlanex16(__float_as_int(x), __float_as_int(x),
                                       0x76543210, 0xFEDCBA98, false, false);
  return __int_as_float(v);
#else
  return __shfl_xor(x, 16, 32);
#endif
}

// value from lane ^ D within each 16-lane half (D = 1,2,4,8)
template <int D>
DEV float swapin16(float x)
{
#ifdef HAVE_PERMLANE
  constexpr unsigned lo = (D == 1) ? 0x67452301u : (D == 2) ? 0x54761032u
                        : (D == 4) ? 0x32107654u : 0xFEDCBA98u;
  constexpr unsigned hi = (D == 1) ? 0xEFCDAB89u : (D == 2) ? 0xDCFE98BAu
                        : (D == 4) ? 0xBA98FEDCu : 0x76543210u;
  int v = __builtin_amdgcn_permlane16(__float_as_int(x), __float_as_int(x),
                                      (int)lo, (int)hi, false, false);
  return __int_as_float(v);
#else
  return __shfl_xor(x, D, 32);
#endif
}

// Sum over m=0..15 of P[n][m] for this lane's n (lane = n + 16*h).
// Per-lane regs p[r] hold P[n][m], m = 4*(r>>1) + (r&1) + 2*h  == WMMA f32 A-operand layout.
DEV float rowsum16(const float (&p)[8], int lane)
{
#ifdef HAVE_WMMA4
  v8f acc = {0.f, 0.f, 0.f, 0.f, 0.f, 0.f, 0.f, 0.f};
  v2f ones; ones[0] = 1.f; ones[1] = 1.f;
#pragma unroll
  for (int c = 0; c < 4; ++c) {
    v2f a; a[0] = p[2 * c]; a[1] = p[2 * c + 1];
    // D = A(16x4 chunk of P) * Ones(4x16) + C  -> row sums accumulate over chunks
    acc = __builtin_amdgcn_wmma_f32_16x16x4_f32(false, a, false, ones,
                                                (short)0, acc, false, false);
  }
  // D layout: vgpr v at lane l holds rowsum(v + 8*(l>=16)). Lane wants row n = l&15.
  const int  idx    = lane & 7;
  const bool useAcc = (((lane >> 3) & 1) == ((lane >> 4) & 1));
  float r = 0.f;
#pragma unroll
  for (int v = 0; v < 8; ++v) {
    float o    = swapx16(acc[v]);
    float cand = useAcc ? acc[v] : o;
    r = (idx == v) ? cand : r;
  }
  return r;
#else
  float s = ((p[0] + p[1]) + (p[2] + p[3])) + ((p[4] + p[5]) + (p[6] + p[7]));
  s += swapx16(s);
  return s;
#endif
}

// Sum over n=0..15 (xor butterfly within each 16-lane half, pure VALU).
DEV float nsum16(float x)
{
  x += swapin16<1>(x);
  x += swapin16<2>(x);
  x += swapin16<4>(x);
  x += swapin16<8>(x);
  return x;
}

// ---------------------------------------------------------------------------
// Pass 0: K-sums of the initial (broadcast) state, layer-0 alpha/beta.
// Wave = one (b,k); block = 8 consecutive k of one b.
// ---------------------------------------------------------------------------
__global__ __launch_bounds__(256)
void bigabp_pass0(const float* __restrict__ Hest, const float* __restrict__ Xest,
                  const float* __restrict__ vX0,  const float* __restrict__ vH0,
                  const float* __restrict__ Yc,   const float* __restrict__ N0c,
                  const float* __restrict__ alphas, const float* __restrict__ betas,
                  float* __restrict__ ws)
{
  __shared__ float sm[8 * 256 * 3];
  const int tid = threadIdx.x, w = tid >> 5, lane = tid & 31;
  const int blk = blockIdx.x, b = blk >> 5, chunk = blk & 31;
  const int k = chunk * 8 + w;
  const int n = lane & 15, h = lane >> 4;
  const int bn = b * 16 + n;

  const float a0 = alphas[0], b0 = betas[0];
  const float mk = (((b & 15) < 2) && (k < 32)) ? 0.f : 1.f;
  const float m2 = a0 * (1.f - mk) + b0 * mk;
  const float Yr = Yc[(bn * 256 + k) * 2], Yi = Yc[(bn * 256 + k) * 2 + 1];
  const float n0 = N0c[bn * 256 + k];

  int mIdx[8];
#pragma unroll
  for (int r = 0; r < 8; ++r) mIdx[r] = 4 * (r >> 1) + (r & 1) + 2 * h;

  float Hr8[8], Hi8[8], Xr8[8], Xi8[8], vX8[8], vH8[8];
#pragma unroll
  for (int r = 0; r < 8; ++r) {
    const int m = mIdx[r];
    const float* hp = Hest + (long)(bn * 16 + m) * 2;
    Hr8[r] = hp[0]; Hi8[r] = hp[1];
    const long xo = (long)(b * 16 + m) * 256 + k;
    const float* xp = Xest + xo * 2;
    Xr8[r] = xp[0]; Xi8[r] = xp[1];
    vX8[r] = vX0[xo];
    vH8[r] = vH0[bn * 16 + m];
  }

  float HXr[8], HXi[8], tp[8];
#pragma unroll
  for (int r = 0; r < 8; ++r) {
    HXr[r] = Hr8[r] * Xr8[r] - Hi8[r] * Xi8[r];
    HXi[r] = Hr8[r] * Xi8[r] + Hi8[r] * Xr8[r];
    const float h2 = Hr8[r] * Hr8[r] + Hi8[r] * Hi8[r];
    const float x2 = Xr8[r] * Xr8[r] + Xi8[r] * Xi8[r];
    tp[r] = h2 * vX8[r] + vH8[r] * (x2 + vX8[r]);
  }
  const float sR = rowsum16(HXr, lane), sI = rowsum16(HXi, lane), sT = rowsum16(tp, lane);

#pragma unroll
  for (int r = 0; r < 8; ++r) {
    const float eR = Yr - sR + HXr[r];
    const float eI = Yi - sI + HXi[r];
    const float xy = sT - tp[r] + n0;
    const float inv = m2 / (xy + vX8[r]);                 // xi_h = xi_y + PHI*vX, PHI=1
    const float x2 = Xr8[r] * Xr8[r] + Xi8[r] * Xi8[r];
    float* q = sm + (w * 256 + (n * 16 + mIdx[r])) * 3;
    q[0] = inv * x2;                                      // vt2 = m2*|X|^2/xi_h
    q[1] = inv * (Xr8[r] * eR + Xi8[r] * eI);             // te2 = m2*conj(X)*err/xi_h
    q[2] = inv * (Xr8[r] * eI - Xi8[r] * eR);
  }
  __syncthreads();
  float* part = ws + OFF_PART + (long)(b * 32 + chunk) * 768;
  for (int i = tid; i < 768; i += 256) {
    float s = 0.f;
#pragma unroll
    for (int w2 = 0; w2 < 8; ++w2) s += sm[w2 * 768 + i];
    part[i] = s;                                          // deterministic fixed-order sum
  }
}

// ---------------------------------------------------------------------------
// Deterministic K-sum finish: Ksum[b][n][m][q] = sum over 32 chunks.
// ---------------------------------------------------------------------------
__global__ __launch_bounds__(256)
void bigabp_reduce(const float* __restrict__ part, float* __restrict__ ks)
{
  const int i = blockIdx.x * 256 + threadIdx.x;
  if (i >= (int)KS_LEN) return;
  const int b = i / 768, rem = i % 768;
  const float* p = part + (long)b * 32 * 768 + rem;
  float s = 0.f;
#pragma unroll
  for (int c = 0; c < 32; ++c) s += p[(long)c * 768];
  ks[i] = s;
}

// ---------------------------------------------------------------------------
// One fused _iter layer: full in-place state update + next layer's K-sum partials.
// ---------------------------------------------------------------------------
template <bool FIRST>
__global__ __launch_bounds__(256)
void bigabp_iter(const float* __restrict__ Hest, const float* __restrict__ Xest,
                 const float* __restrict__ vX0,  const float* __restrict__ vH0,
                 const float* __restrict__ Yc,   const float* __restrict__ N0c,
                 const float* __restrict__ alphas, const float* __restrict__ betas,
                 const float* __restrict__ gammas, const float* __restrict__ etas,
                 float* __restrict__ ws, int layer)
{
  __shared__ float sm[8 * 256 * 3];
  __shared__ float ksm[768];
  const int tid = threadIdx.x, w = tid >> 5, lane = tid & 31;
  const int blk = blockIdx.x, b = blk >> 5, chunk = blk & 31;
  const int k = chunk * 8 + w;
  const int n = lane & 15, h = lane >> 4;
  const int bn = b * 16 + n;

  float* Hr_s = ws + OFF_HR; float* Hi_s = ws + OFF_HI;
  float* Xr_s = ws + OFF_XR; float* Xi_s = ws + OFF_XI;
  float* vXs  = ws + OFF_VX; float* vHs  = ws + OFF_VH;
  const float* Kp = ws + OFF_KS + (long)b * 768;
  const long slab = (long)(b * 256 + k) * 256;            // state: [b][k][n][m]

  __builtin_prefetch(Kp, 0, 1);                           // global_prefetch_b8

  // stage this batch's K-sum row (768 floats) into LDS, coalesced
  for (int i = tid; i < 768; i += 256) ksm[i] = Kp[i];

  const float a_i = alphas[layer],     b_i = betas[layer];
  const float a_n = alphas[layer + 1], b_n = betas[layer + 1];
  const float g_i = gammas[layer];
  const float eta = etas[layer];
  const float mk  = (((b & 15) < 2) && (k < 32)) ? 0.f : 1.f;
  const float m2  = a_i * (1.f - mk) + b_i * mk;
  const float m2n = a_n * (1.f - mk) + b_n * mk;
  const float Yr = Yc[(bn * 256 + k) * 2], Yi = Yc[(bn * 256 + k) * 2 + 1];
  const float n0 = N0c[bn * 256 + k];

  int mIdx[8];
#pragma unroll
  for (int r = 0; r < 8; ++r) mIdx[r] = 4 * (r >> 1) + (r & 1) + 2 * h;

  float Hr8[8], Hi8[8], Xr8[8], Xi8[8], vX8[8], vH8[8];
  if (FIRST) {
#pragma unroll
    for (int r = 0; r < 8; ++r) {
      const int m = mIdx[r];
      const float* hp = Hest + (long)(bn * 16 + m) * 2;
      Hr8[r] = hp[0]; Hi8[r] = hp[1];
      const long xo = (long)(b * 16 + m) * 256 + k;
      const float* xp = Xest + xo * 2;
      Xr8[r] = xp[0]; Xi8[r] = xp[1];
      vX8[r] = vX0[xo];
      vH8[r] = vH0[bn * 16 + m];
    }
  } else {
#pragma unroll
    for (int c = 0; c < 4; ++c) {
      const long off = slab + n * 16 + 4 * c + 2 * h;
      float2 t;
      t = *(const float2*)(Hr_s + off); Hr8[2*c] = t.x; Hr8[2*c+1] = t.y;
      t = *(const float2*)(Hi_s + off); Hi8[2*c] = t.x; Hi8[2*c+1] = t.y;
      t = *(const float2*)(Xr_s + off); Xr8[2*c] = t.x; Xr8[2*c+1] = t.y;
      t = *(const float2*)(Xi_s + off); Xi8[2*c] = t.x; Xi8[2*c+1] = t.y;
      t = *(const float2*)(vXs  + off); vX8[2*c] = t.x; vX8[2*c+1] = t.y;
      t = *(const float2*)(vHs  + off); vH8[2*c] = t.x; vH8[2*c+1] = t.y;
    }
  }

  // ---- round 1: err / xi from old state -------------------------------------
  float HXr[8], HXi[8], tp[8];
#pragma unroll
  for (int r = 0; r < 8; ++r) {
    HXr[r] = Hr8[r] * Xr8[r] - Hi8[r] * Xi8[r];
    HXi[r] = Hr8[r] * Xi8[r] + Hi8[r] * Xr8[r];
    const float h2 = Hr8[r] * Hr8[r] + Hi8[r] * Hi8[r];
    const float x2 = Xr8[r] * Xr8[r] + Xi8[r] * Xi8[r];
    tp[r] = h2 * vX8[r] + vH8[r] * (x2 + vX8[r]);
  }
  const float sR = rowsum16(HXr, lane), sI = rowsum16(HXi, lane), sT = rowsum16(tp, lane);

  float eR[8], eI[8], xiy[8];
#pragma unroll
  for (int r = 0; r < 8; ++r) {
    eR[r]  = Yr - sR + HXr[r];
    eI[r]  = Yi - sI + HXi[r];
    xiy[r] = sT - tp[r] + n0;
  }

  // ---- H-path part 1 (uses OLD X) -------------------------------------------
  float vt2[8], t2R[8], t2I[8];
#pragma unroll
  for (int r = 0; r < 8; ++r) {
    const float inv = m2 / (xiy[r] + vX8[r]);
    const float x2 = Xr8[r] * Xr8[r] + Xi8[r] * Xi8[r];
    vt2[r] = inv * x2;
    t2R[r] = inv * (Xr8[r] * eR[r] + Xi8[r] * eI[r]);
    t2I[r] = inv * (Xr8[r] * eI[r] - Xi8[r] * eR[r]);
  }

  // ---- X-path: N-sums (minus self), demod_soft, damped update ---------------
#pragma unroll
  for (int r = 0; r < 8; ++r) {
    const float inv = 1.f / (xiy[r] + vH8[r]);            // xi_x = xi_y + vH
    const float vt  = (Hr8[r] * Hr8[r] + Hi8[r] * Hi8[r]) * inv;
    const float teR = (Hr8[r] * eR[r] + Hi8[r] * eI[r]) * inv;
    const float teI = (Hr8[r] * eI[r] - Hi8[r] * eR[r]) * inv;
    const float Svt = nsum16(vt), StR = nsum16(teR), StI = nsum16(teI);
    const float var = 1.f / (Svt - vt);
    const float estR = (StR - teR) * var, estI = (StI - teI) * var;
    const float c2 = 2.f * Sq / g_i;
    const float mr = tanhf(c2 * estR), mi = tanhf(c2 * estI);
    const float vp = 1.f - Sq * Sq * (mr * mr + mi * mi);
    Xr8[r] += eta * (Sq * mr - Xr8[r]) * mk;
    Xi8[r] += eta * (Sq * mi - Xi8[r]) * mk;
    vX8[r] += eta * (vp     - vX8[r]) * mk;
  }

  // ---- H-path part 2: K-sums (minus self) from LDS, demod_H, damped update --
  __syncthreads();                                        // ksm ready
#pragma unroll
  for (int r = 0; r < 8; ++r) {
    const int km = (n * 16 + mIdx[r]) * 3;
    const float var2  = 1.f / (ksm[km] - vt2[r]);
    const float est2R = (ksm[km + 1] - t2R[r]) * var2;
    const float est2I = (ksm[km + 2] - t2I[r]) * var2;
    const float g = 1.f / (1.f + var2);                   // PHI/(PHI+var), PHI=1
    Hr8[r] += eta * (g * est2R - Hr8[r]);
    Hi8[r] += eta * (g * est2I - Hi8[r]);
    vH8[r] += eta * (g * var2  - vH8[r]);
  }

  // ---- store updated state (coalesced float2 within the wave's 1KB slab) ----
#pragma unroll
  for (int c = 0; c < 4; ++c) {
    const long off = slab + n * 16 + 4 * c + 2 * h;
    *(float2*)(Hr_s + off) = make_float2(Hr8[2*c], Hr8[2*c+1]);
    *(float2*)(Hi_s + off) = make_float2(Hi8[2*c], Hi8[2*c+1]);
    *(float2*)(Xr_s + off) = make_float2(Xr8[2*c], Xr8[2*c+1]);
    *(float2*)(Xi_s + off) = make_float2(Xi8[2*c], Xi8[2*c+1]);
    *(float2*)(vXs  + off) = make_float2(vX8[2*c], vX8[2*c+1]);
    *(float2*)(vHs  + off) = make_float2(vH8[2*c], vH8[2*c+1]);
  }

  // ---- round 2: next layer's K-sum partials from UPDATED state --------------
#pragma unroll
  for (int r = 0; r < 8; ++r) {
    HXr[r] = Hr8[r] * Xr8[r] - Hi8[r] * Xi8[r];
    HXi[r] = Hr8[r] * Xi8[r] + Hi8[r] * Xr8[r];
    const float h2 = Hr8[r] * Hr8[r] + Hi8[r] * Hi8[r];
    const float x2 = Xr8[r] * Xr8[r] + Xi8[r] * Xi8[r];
    tp[r] = h2 * vX8[r] + vH8[r] * (x2 + vX8[r]);
  }
  const float s2R = rowsum16(HXr, lane), s2I = rowsum16(HXi, lane), s2T = rowsum16(tp, lane);
#pragma unroll
  for (int r = 0; r < 8; ++r) {
    const float er = Yr - s2R + HXr[r];
    const float ei = Yi - s2I + HXi[r];
    const float xy = s2T - tp[r] + n0;
    const float inv = m2n / (xy + vX8[r]);
    const float x2 = Xr8[r] * Xr8[r] + Xi8[r] * Xi8[r];
    float* q = sm + (w * 256 + (n * 16 + mIdx[r])) * 3;
    q[0] = inv * x2;
    q[1] = inv * (Xr8[r] * er + Xi8[r] * ei);
    q[2] = inv * (Xr8[r] * ei - Xi8[r] * er);
  }
  __syncthreads();
  float* part = ws + OFF_PART + (long)(b * 32 + chunk) * 768;
  for (int i = tid; i < 768; i += 256) {
    float s = 0.f;
#pragma unroll
    for (int w2 = 0; w2 < 8; ++w2) s += sm[w2 * 768 + i];
    part[i] = s;
  }
}

// ---------------------------------------------------------------------------
// _last, X outputs: full N-sums (no minus-self), demod_hard -> Xp, vXp.
// ---------------------------------------------------------------------------
__global__ __launch_bounds__(256)
void bigabp_lastX(const float* __restrict__ ws, const float* __restrict__ Yc,
                  const float* __restrict__ N0c, float* __restrict__ outp)
{
  const int tid = threadIdx.x, w = tid >> 5, lane = tid & 31;
  const int blk = blockIdx.x, b = blk >> 5, chunk = blk & 31;
  const int k = chunk * 8 + w;
  const int n = lane & 15, h = lane >> 4;
  const int bn = b * 16 + n;

  const float* Hr_s = ws + OFF_HR; const float* Hi_s = ws + OFF_HI;
  const float* Xr_s = ws + OFF_XR; const float* Xi_s = ws + OFF_XI;
  const float* vXs  = ws + OFF_VX; const float* vHs  = ws + OFF_VH;
  const long slab = (long)(b * 256 + k) * 256;

  const float mk = (((b & 15) < 2) && (k < 32)) ? 0.f : 1.f;
  const float Yr = Yc[(bn * 256 + k) * 2], Yi = Yc[(bn * 256 + k) * 2 + 1];
  const float n0 = N0c[bn * 256 + k];

  int mIdx[8];
#pragma unroll
  for (int r = 0; r < 8; ++r) mIdx[r] = 4 * (r >> 1) + (r & 1) + 2 * h;

  float Hr8[8], Hi8[8], Xr8[8], Xi8[8], vX8[8], vH8[8];
#pragma unroll
  for (int c = 0; c < 4; ++c) {
    const long off = slab + n * 16 + 4 * c + 2 * h;
    float2 t;
    t = *(const float2*)(Hr_s + off); Hr8[2*c] = t.x; Hr8[2*c+1] = t.y;
    t = *(const float2*)(Hi_s + off); Hi8[2*c] = t.x; Hi8[2*c+1] = t.y;
    t = *(const float2*)(Xr_s + off); Xr8[2*c] = t.x; Xr8[2*c+1] = t.y;
    t = *(const float2*)(Xi_s + off); Xi8[2*c] = t.x; Xi8[2*c+1] = t.y;
    t = *(const float2*)(vXs  + off); vX8[2*c] = t.x; vX8[2*c+1] = t.y;
    t = *(const float2*)(vHs  + off); vH8[2*c] = t.x; vH8[2*c+1] = t.y;
  }

  float HXr[8], HXi[8], tp[8];
#pragma unroll
  for (int r = 0; r < 8; ++r) {
    HXr[r] = Hr8[r] * Xr8[r] - Hi8[r] * Xi8[r];
    HXi[r] = Hr8[r] * Xi8[r] + Hi8[r] * Xr8[r];
    const float h2 = Hr8[r] * Hr8[r] + Hi8[r] * Hi8[r];
    const float x2 = Xr8[r] * Xr8[r] + Xi8[r] * Xi8[r];
    tp[r] = h2 * vX8[r] + vH8[r] * (x2 + vX8[r]);
  }
  const float sR = rowsum16(HXr, lane), sI = rowsum16(HXi, lane), sT = rowsum16(tp, lane);

#pragma unroll
  for (int r = 0; r < 8; ++r) {
    const float eR = Yr - sR + HXr[r];
    const float eI = Yi - sI + HXi[r];
    const float xy = sT - tp[r] + n0;
    const float inv = 1.f / (xy + vH8[r]);                // xi_x
    const float vt  = (Hr8[r] * Hr8[r] + Hi8[r] * Hi8[r]) * inv;
    const float teR = (Hr8[r] * eR + Hi8[r] * eI) * inv;
    const float teI = (Hr8[r] * eI - Hi8[r] * eR) * inv;
    const float Svt = nsum16(vt), StR = nsum16(teR), StI = nsum16(teI);
    const float var = 1.f / Svt;                          // no minus-self in _last
    const float sr = sgnf(StR * var), si = sgnf(StI * var);
    if (n == 0) {                                         // lanes 0 and 16 write 8 m's each
      const long om = (long)(b * 16 + mIdx[r]) * 256 + k;
      outp[OX  + om * 2]     = Sq * sr * mk;
      outp[OX  + om * 2 + 1] = Sq * si * mk;
      outp[OVX + om]         = (1.f - Sq * Sq * (sr * sr + si * si)) * mk;
    }
  }
}

// ---------------------------------------------------------------------------
// _last, H outputs: purely from final K-sums.
// ---------------------------------------------------------------------------
__global__ __launch_bounds__(256)
void bigabp_lastH(const float* __restrict__ ks, float* __restrict__ outp)
{
  const int i = blockIdx.x * 256 + threadIdx.x;           // i = b*256 + n*16 + m
  if (i >= Bc * Nc * Mc) return;
  const float var = 1.f / ks[i * 3];
  const float g = 1.f / (1.f + var);
  outp[i * 2]     = g * ks[i * 3 + 1] * var;
  outp[i * 2 + 1] = g * ks[i * 3 + 2] * var;
  outp[OVH + i]   = g * var;
}

} // namespace

extern "C" void kernel_launch(void* const* d_in, const int* in_sizes, int n_in,
                              void* d_out, int out_size, void* d_ws, size_t ws_size,
                              hipStream_t stream)
{
  (void)in_sizes; (void)n_in; (void)out_size; (void)ws_size;
  const float* Hest = (const float*)d_in[0];
  const float* Xest = (const float*)d_in[1];
  const float* vX0  = (const float*)d_in[2];
  const float* vH0  = (const float*)d_in[3];
  const float* Yc   = (const float*)d_in[4];
  const float* N0c  = (const float*)d_in[5];
  const float* al   = (const float*)d_in[6];
  const float* be   = (const float*)d_in[7];
  const float* ga   = (const float*)d_in[8];
  const float* et   = (const float*)d_in[9];
  float* outp = (float*)d_out;
  float* ws   = (float*)d_ws;

  bigabp_pass0<<<2048, 256, 0, stream>>>(Hest, Xest, vX0, vH0, Yc, N0c, al, be, ws);
  bigabp_reduce<<<192, 256, 0, stream>>>(ws + OFF_PART, ws + OFF_KS);
  bigabp_iter<true ><<<2048, 256, 0, stream>>>(Hest, Xest, vX0, vH0, Yc, N0c, al, be, ga, et, ws, 0);
  bigabp_reduce<<<192, 256, 0, stream>>>(ws + OFF_PART, ws + OFF_KS);
  bigabp_iter<false><<<2048, 256, 0, stream>>>(Hest, Xest, vX0, vH0, Yc, N0c, al, be, ga, et, ws, 1);
  bigabp_reduce<<<192, 256, 0, stream>>>(ws + OFF_PART, ws + OFF_KS);
  bigabp_iter<false><<<2048, 256, 0, stream>>>(Hest, Xest, vX0, vH0, Yc, N0c, al, be, ga, et, ws, 2);
  bigabp_reduce<<<192, 256, 0, stream>>>(ws + OFF_PART, ws + OFF_KS);
  bigabp_lastX<<<2048, 256, 0, stream>>>(ws, Yc, N0c, outp);
  bigabp_lastH<<<64, 256, 0, stream>>>(ws + OFF_KS, outp);
}